// GAE_36859409334558
// MI455X (gfx1250) — compile-verified
//
#include <hip/hip_runtime.h>
#include <math.h>

// ---------------------------------------------------------------------------
// GAE on MI455X (gfx1250, wave32).
//
// adv[t] = delta[t] + (G*L)*m[t]*adv[t+1],  delta = r + G*nv*m - v
// ret[t] = r[t]     +  G   *m[t]*ret[t+1]
// advantages globally normalized by mean / unbiased std (+1e-8).
//
// Per 16-row x 16-time tile the recurrence is solved in closed form:
//   adv[i] = ( S[i] + q16*carry ) * rcp(q[i]),  S[i] = sum_{j>=i} delta[j]*q[j]
// where q[j] = prod_{u<j} c[u].  S is a matmul with the CONSTANT
// upper-triangular matrix Tri[i][j] = (j >= i)  ->  V_WMMA_F32_16X16X4_F32 x4.
// mask==0 is clamped to 2^-40 (a power of two) so reset segments cancel
// exactly in the q[j]/q[i] ratio (exact for <=3 resets per 16-tile).
//
// Input staging: Tensor Data Mover.  One tensor_load_to_lds per input per
// tile (16x16 f32, tensor_dim0_stride=T, LDS padding 1 dword / 16 dwords ->
// stride-17 conflict-free rows), double buffered, TENSORcnt-synchronized, so
// HBM latency overlaps the WMMA chain of the previous tile.
// ---------------------------------------------------------------------------

typedef __attribute__((ext_vector_type(2)))  float    v2f;
typedef __attribute__((ext_vector_type(8)))  float    v8f;
typedef __attribute__((ext_vector_type(4)))  unsigned v4u;
typedef __attribute__((ext_vector_type(4)))  int      v4i;
typedef __attribute__((ext_vector_type(8)))  int      v8i;

#define GAE_G  0.99f
#define GAE_GL (0.99f * 0.95f)
#define GAE_CMIN 0x1p-40f   // power-of-two clamp: exact cancellation in ratios

#if defined(__has_builtin)
#if __has_builtin(__builtin_amdgcn_tensor_load_to_lds) && \
    __has_builtin(__builtin_amdgcn_s_wait_tensorcnt)
#define GAE_TDM 1
#endif
#endif

#ifdef GAE_TDM
// Issue one TDM load of a 16x16 f32 tile (row stride T elements in memory)
// into LDS at lds_byte_off, padded to 17-float rows.  Async: TENSORcnt.
__device__ inline void tdm_tile(const float* gptr, unsigned lds_byte_off, int T) {
  unsigned long long ga = (unsigned long long)(uintptr_t)gptr;   // 57-bit byte addr
  v4u g0;
  g0[0] = 1u;                                         // count=1 (valid, user mode)
  g0[1] = lds_byte_off;                               // D#.lds_addr
  g0[2] = (unsigned)(ga & 0xFFFFFFFFu);               // global_addr[31:0]
  g0[3] = (unsigned)((ga >> 32) & 0x01FFFFFFu)        // global_addr[56:32]
        | 0x80000000u;                                // type=2 ("image")
  v8i g1;
  g1[0] = (int)0x00D20000;      // data_size=2(4B) | pad_enable | pad_interval=3(16dw)
                                // pad_amount=0(1dw); workgroup_mask=0 (not in cluster)
  g1[1] = (int)(16u << 16);     // tensor_dim0 = 16 (tile-relative, never OOB)
  g1[2] = (int)(16u << 16);     // tensor_dim0_hi=0 | tensor_dim1 = 16
  g1[3] = (int)(16u << 16);     // tensor_dim1_hi=0 | tile_dim0 = 16
  g1[4] = 16;                   // tile_dim1 = 16, tile_dim2 = 0 (2D tile)
  g1[5] = T;                    // tensor_dim0_stride = T elements
  g1[6] = 0;                    // stride0_hi=0 | tensor_dim1_stride lo (unused)
  g1[7] = 0;
  v4i z4 = {0, 0, 0, 0};
#if __clang_major__ >= 23
  v8i z8 = {0, 0, 0, 0, 0, 0, 0, 0};
  __builtin_amdgcn_tensor_load_to_lds(g0, g1, z4, z4, z8, 0);   // 6-arg toolchain
#else
  __builtin_amdgcn_tensor_load_to_lds(g0, g1, z4, z4, 0);       // ROCm 7.2 (clang-22)
#endif
}
#endif

__device__ inline void unpack8(float4 a, float4 b, float* d) {
  d[0]=a.x; d[1]=a.y; d[2]=a.z; d[3]=a.w;
  d[4]=b.x; d[5]=b.y; d[6]=b.z; d[7]=b.w;
}

__global__ void __launch_bounds__(32)
gae_scan_kernel(const float* __restrict__ rew, const float* __restrict__ val,
                const float* __restrict__ nxt, const float* __restrict__ msk,
                float* __restrict__ outAdv, float* __restrict__ outRet,
                double* __restrict__ partials, int T)
{
  const int lane = threadIdx.x & 31;
  const int b    = lane & 15;        // row within tile == WMMA N index
  const int hi   = lane >> 4;        // half-wave select
  const int jb   = hi * 8;           // local time base (C/D layout: M = vgpr + 8*hi)
  const long long rowBase = (long long)blockIdx.x * 16;
  const long long rowOff  = (rowBase + b) * (long long)T;

  __shared__ float YA[16][17];       // y = delta*q, [time j][row b], padded
  __shared__ float YR[16][17];       // y = r*q
#ifdef GAE_TDM
  __shared__ float IN[2][4][16 * 17];  // double-buffered TDM tiles, stride-17 rows
#endif

  // Constant Tri operand, A-layout (16x4 f32): v0: K = kk*4 + 2*hi, v1: K+1
  v2f triA[4];
  for (int kk = 0; kk < 4; ++kk) {
    int k0 = kk * 4 + hi * 2;
    triA[kk][0] = (k0     >= b) ? 1.0f : 0.0f;
    triA[kk][1] = (k0 + 1 >= b) ? 1.0f : 0.0f;
  }

  double sumA = 0.0, sumQ = 0.0;
  float carryA = 0.0f, carryR = 0.0f;
  int cur = 0;

#ifdef GAE_TDM
  {  // prologue: DMA first (latest-time) tile into buffer 0
    const long long g = rowBase * (long long)T + (T - 16);
    tdm_tile(rew + g, (unsigned)(uintptr_t)&IN[0][0][0], T);
    tdm_tile(val + g, (unsigned)(uintptr_t)&IN[0][1][0], T);
    tdm_tile(nxt + g, (unsigned)(uintptr_t)&IN[0][2][0], T);
    tdm_tile(msk + g, (unsigned)(uintptr_t)&IN[0][3][0], T);
  }
#endif

  for (int t0 = T - 16; t0 >= 0; t0 -= 16) {
    const long long off = rowOff + t0 + jb;
    float r[8], v_[8], n_[8], m_[8];

#ifdef GAE_TDM
    if (t0 >= 16) {
      // WAR guard: prior LDS reads of the recycled buffer must retire before
      // TDM starts writing it; also a compiler memory fence.
      asm volatile("s_wait_dscnt 0x0" ::: "memory");
      const long long g = rowBase * (long long)T + (t0 - 16);
      const int nb = cur ^ 1;
      tdm_tile(rew + g, (unsigned)(uintptr_t)&IN[nb][0][0], T);
      tdm_tile(val + g, (unsigned)(uintptr_t)&IN[nb][1][0], T);
      tdm_tile(nxt + g, (unsigned)(uintptr_t)&IN[nb][2][0], T);
      tdm_tile(msk + g, (unsigned)(uintptr_t)&IN[nb][3][0], T);
      __builtin_amdgcn_s_wait_tensorcnt(4);   // current tile's 4 DMAs done
    } else {
      __builtin_amdgcn_s_wait_tensorcnt(0);   // final tile: drain
    }
    asm volatile("" ::: "memory");
    const int ib = b * 17 + jb;
    #pragma unroll
    for (int k = 0; k < 8; ++k) {
      r[k]  = IN[cur][0][ib + k];
      v_[k] = IN[cur][1][ib + k];
      n_[k] = IN[cur][2][ib + k];
      m_[k] = IN[cur][3][ib + k];
    }
#else
    float4 r0 = *(const float4*)(rew + off); float4 r1 = *(const float4*)(rew + off + 4);
    float4 v0 = *(const float4*)(val + off); float4 v1 = *(const float4*)(val + off + 4);
    float4 n0 = *(const float4*)(nxt + off); float4 n1 = *(const float4*)(nxt + off + 4);
    float4 m0 = *(const float4*)(msk + off); float4 m1 = *(const float4*)(msk + off + 4);
    if (t0 >= 16) {
      __builtin_prefetch(rew + off - 16, 0, 1);
      __builtin_prefetch(val + off - 16, 0, 1);
      __builtin_prefetch(nxt + off - 16, 0, 1);
      __builtin_prefetch(msk + off - 16, 0, 1);
    }
    unpack8(r0, r1, r); unpack8(v0, v1, v_); unpack8(n0, n1, n_); unpack8(m0, m1, m_);
#endif

    float delta[8], cA[8], cR[8];
    #pragma unroll
    for (int k = 0; k < 8; ++k) {
      float mm = m_[k];
      delta[k] = r[k] + GAE_G * n_[k] * mm - v_[k];
      cA[k] = fmaxf(GAE_GL * mm, GAE_CMIN);
      cR[k] = fmaxf(GAE_G  * mm, GAE_CMIN);
    }

    // half-lane prefix products, stitched across halves with shfl_xor(16)
    float pA[8], pR[8];
    pA[0] = 1.0f; pR[0] = 1.0f;
    #pragma unroll
    for (int k = 1; k < 8; ++k) { pA[k] = pA[k-1]*cA[k-1]; pR[k] = pR[k-1]*cR[k-1]; }
    float totA = pA[7]*cA[7], totR = pR[7]*cR[7];
    float ptA = __shfl_xor(totA, 16);
    float ptR = __shfl_xor(totR, 16);
    float fA = hi ? ptA : 1.0f;
    float fR = hi ? ptR : 1.0f;
    float q16A = totA * ptA;             // identical on both halves
    float q16R = totR * ptR;

    float qA[8], qR[8];
    #pragma unroll
    for (int k = 0; k < 8; ++k) { qA[k] = pA[k]*fA; qR[k] = pR[k]*fR; }

    #pragma unroll
    for (int k = 0; k < 8; ++k) {
      YA[jb + k][b] = delta[k] * qA[k];
      YR[jb + k][b] = r[k]     * qR[k];
    }
    __syncthreads();   // single-wave block: barrier is S_NOP, orders DS ops

    v8f SA = {0.f,0.f,0.f,0.f,0.f,0.f,0.f,0.f};
    v8f SR = {0.f,0.f,0.f,0.f,0.f,0.f,0.f,0.f};
    #pragma unroll
    for (int kk = 0; kk < 4; ++kk) {
      int k0 = kk * 4 + hi * 2;          // B layout mirrors A: v0:K=k0, v1:K=k0+1
      v2f bA, bR;
      bA[0] = YA[k0][b]; bA[1] = YA[k0+1][b];
      bR[0] = YR[k0][b]; bR[1] = YR[k0+1][b];
      SA = __builtin_amdgcn_wmma_f32_16x16x4_f32(false, triA[kk], false, bA,
                                                 (short)0, SA, false, false);
      SR = __builtin_amdgcn_wmma_f32_16x16x4_f32(false, triA[kk], false, bR,
                                                 (short)0, SR, false, false);
    }
    __syncthreads();   // LDS reads done before next tile overwrites

    // q in [0.94^15, 1] per tile: one v_rcp_f32 (+~1 ulp) replaces full IEEE div
    float adv[8], ret[8];
    #pragma unroll
    for (int k = 0; k < 8; ++k) {
      adv[k] = (SA[k] + q16A * carryA) * __builtin_amdgcn_rcpf(qA[k]);
      ret[k] = (SR[k] + q16R * carryR) * __builtin_amdgcn_rcpf(qR[k]);
    }
    carryA = __shfl(adv[0], b);          // broadcast row b's t0 value to both halves
    carryR = __shfl(ret[0], b);

    float4 oa0 = make_float4(adv[0], adv[1], adv[2], adv[3]);
    float4 oa1 = make_float4(adv[4], adv[5], adv[6], adv[7]);
    float4 or0 = make_float4(ret[0], ret[1], ret[2], ret[3]);
    float4 or1 = make_float4(ret[4], ret[5], ret[6], ret[7]);
    *(float4*)(outAdv + off)     = oa0;  *(float4*)(outAdv + off + 4) = oa1;
    *(float4*)(outRet + off)     = or0;  *(float4*)(outRet + off + 4) = or1;

    #pragma unroll
    for (int k = 0; k < 8; ++k) {
      sumA += (double)adv[k];
      sumQ += (double)adv[k] * (double)adv[k];
    }
    cur ^= 1;
  }

  // deterministic wave reduction of partial sums
  #pragma unroll
  for (int d = 16; d >= 1; d >>= 1) {
    sumA += __shfl_xor(sumA, d);
    sumQ += __shfl_xor(sumQ, d);
  }
  if (lane == 0) {
    partials[2 * blockIdx.x]     = sumA;
    partials[2 * blockIdx.x + 1] = sumQ;
  }
}

__global__ void __launch_bounds__(256)
gae_reduce_kernel(const double* __restrict__ partials, int n,
                  double* __restrict__ stats, double N)
{
  __shared__ double ss[256], sq[256];
  int t = threadIdx.x;
  double a = 0.0, q = 0.0;
  for (int i = t; i < n; i += 256) { a += partials[2*i]; q += partials[2*i+1]; }
  ss[t] = a; sq[t] = q;
  __syncthreads();
  for (int s = 128; s > 0; s >>= 1) {
    if (t < s) { ss[t] += ss[t+s]; sq[t] += sq[t+s]; }
    __syncthreads();
  }
  if (t == 0) {
    double mean = ss[0] / N;
    double var  = (sq[0] - N * mean * mean) / (N - 1.0);  // ddof=1
    if (var < 0.0) var = 0.0;
    stats[0] = mean;
    stats[1] = 1.0 / (sqrt(var) + 1e-8);
  }
}

__global__ void __launch_bounds__(256)
gae_norm_kernel(float* __restrict__ adv, const double* __restrict__ stats, long long n4)
{
  const float mean  = (float)stats[0];
  const float scale = (float)stats[1];
  long long i      = (long long)blockIdx.x * blockDim.x + threadIdx.x;
  long long stride = (long long)gridDim.x * blockDim.x;
  float4* p = (float4*)adv;
  for (; i < n4; i += stride) {
    float4 x = p[i];
    x.x = (x.x - mean) * scale;
    x.y = (x.y - mean) * scale;
    x.z = (x.z - mean) * scale;
    x.w = (x.w - mean) * scale;
    p[i] = x;
  }
}

extern "C" void kernel_launch(void* const* d_in, const int* in_sizes, int n_in,
                              void* d_out, int out_size, void* d_ws, size_t ws_size,
                              hipStream_t stream) {
  const float* rew = (const float*)d_in[0];
  const float* val = (const float*)d_in[1];
  const float* nxt = (const float*)d_in[2];
  const float* msk = (const float*)d_in[3];

  const long long total = (long long)in_sizes[0];  // B*T
  const int T  = 4096;                             // fixed by reference
  const int Bn = (int)(total / T);
  const int nBlocks = Bn / 16;                     // 16 rows per wave

  float* outAdv = (float*)d_out;
  float* outRet = outAdv + total;

  double* partials = (double*)d_ws;                  // [2 * nBlocks]
  double* stats    = partials + 2 * (size_t)nBlocks; // [2] mean, 1/(std+eps)

  gae_scan_kernel<<<nBlocks, 32, 0, stream>>>(rew, val, nxt, msk,
                                              outAdv, outRet, partials, T);
  gae_reduce_kernel<<<1, 256, 0, stream>>>(partials, nBlocks, stats, (double)total);
  gae_norm_kernel<<<2048, 256, 0, stream>>>(outAdv, stats, total / 4);
}